// PoseGraphNet_91190745629227
// MI455X (gfx1250) — compile-verified
//
#include <hip/hip_runtime.h>

typedef __attribute__((ext_vector_type(2))) float v2f;
typedef __attribute__((ext_vector_type(8))) float v8f;
typedef int v4i_vec __attribute__((vector_size(16)));   // 16-byte int vector

#define NN 100            // nodes
#define KDIM 200          // 2*NN
#define ROWS 128          // rows of x per block (8 waves * 16)
#define XS_STRIDE 204     // LDS row stride for X tile (mult of 4, good banks)
#define WS_STRIDE 202     // LDS row stride for W_eff^T (even, good banks)

#if defined(__gfx1250__) && \
    __has_builtin(__builtin_amdgcn_global_load_async_to_lds_b128) && \
    __has_builtin(__builtin_amdgcn_s_wait_asynccnt)
#define USE_ASYNC_LDS 1
#pragma message("CDNA5: using global_load_async_to_lds_b128 staging path")
#else
#define USE_ASYNC_LDS 0
#pragma message("CDNA5: async-to-LDS builtins unavailable, using load+ds_store staging")
#endif

// ---------------------------------------------------------------------------
// Kernel 1: tiny edge-MLP -> adjacency -> normalized Laplacian -> fused 2x200
// weight W_eff[o][2j+d] = sum_i fc_w[o][2i+d] * dvec[i]*(A[i][j]+I)*dvec[j]
// ---------------------------------------------------------------------------
__global__ void __launch_bounds__(256)
posegraph_prep_kernel(const float* __restrict__ coord,   // [100*100*4]
                      const float* __restrict__ w1,      // [64*4]
                      const float* __restrict__ b1,      // [64]
                      const float* __restrict__ w2,      // [64]
                      const float* __restrict__ b2,      // [1]
                      const float* __restrict__ fc_w,    // [2*200]
                      float* __restrict__ Wc) {          // out: [2*200]
    __shared__ float A[NN * NN];      // 40 KB
    __shared__ float dvec[NN];
    __shared__ float w1s[256];
    __shared__ float b1s[64];
    __shared__ float w2s[64];

    const int tid = threadIdx.x;
    w1s[tid] = w1[tid];               // 256 threads, 256 floats
    if (tid < 64) { b1s[tid] = b1[tid]; w2s[tid] = w2[tid]; }
    const float b2v = b2[0];
    __syncthreads();

    // A[i][j] = tanh(w2 . relu(w1 @ pair + b1) + b2)
    for (int e = tid; e < NN * NN; e += 256) {
        const float c0 = coord[e * 4 + 0];
        const float c1 = coord[e * 4 + 1];
        const float c2 = coord[e * 4 + 2];
        const float c3 = coord[e * 4 + 3];
        float s = b2v;
#pragma unroll
        for (int j = 0; j < 64; ++j) {
            float h = fmaf(w1s[j * 4 + 0], c0,
                      fmaf(w1s[j * 4 + 1], c1,
                      fmaf(w1s[j * 4 + 2], c2,
                      fmaf(w1s[j * 4 + 3], c3, b1s[j]))));
            h = fmaxf(h, 0.0f);
            s = fmaf(w2s[j], h, s);
        }
        A[e] = tanhf(s);
    }
    __syncthreads();

    // dvec[i] = (rowsum(A+I) + 1e-5)^-0.5
    if (tid < NN) {
        float sum = 1.0f;             // identity diagonal
        for (int j = 0; j < NN; ++j) sum += A[tid * NN + j];
        dvec[tid] = rsqrtf(sum + 1e-5f);
    }
    __syncthreads();

    // W_eff[o][k] with k = 2j+d
    for (int idx = tid; idx < 2 * KDIM; idx += 256) {
        const int o = idx / KDIM;
        const int k = idx % KDIM;
        const int j = k >> 1;
        const int d = k & 1;
        const float dj = dvec[j];
        float sum = 0.0f;
        for (int i = 0; i < NN; ++i) {
            float ahat = A[i * NN + j] + (i == j ? 1.0f : 0.0f);
            sum = fmaf(fc_w[o * KDIM + 2 * i + d], dvec[i] * ahat * dj, sum);
        }
        Wc[idx] = sum;
    }
}

// ---------------------------------------------------------------------------
// Kernel 2: out[B,2] = x[B,200] @ W_eff^T  via V_WMMA_F32_16X16X4_F32
// Each block stages 128 rows of x into LDS (async direct-to-LDS if available,
// else coalesced float4 + ds_store), each of the 8 waves computes one 16-row
// tile, K = 200 in 50 WMMA steps of 4.
// ---------------------------------------------------------------------------
__global__ void __launch_bounds__(256)
posegraph_gemm_kernel(const float* __restrict__ X,   // [B*200]
                      const float* __restrict__ Wc,  // [2*200]
                      float* __restrict__ out,       // [B*2]
                      int Brows) {
    extern __shared__ float smem[];
    float* Xs = smem;                            // [ROWS][XS_STRIDE]
    float* Ws = smem + ROWS * XS_STRIDE;         // [16][WS_STRIDE]

    const int tid = threadIdx.x;
    const int row0 = blockIdx.x * ROWS;

    // zero-pad W_eff^T to 16 N-columns (rows 2..15 = 0)
    for (int i = tid; i < 16 * WS_STRIDE; i += 256) Ws[i] = 0.0f;

    // ---- stage 128 x 200 floats of X into LDS ----
#if USE_ASYNC_LDS
    typedef __attribute__((address_space(1))) v4i_vec* gvec_t;
    typedef __attribute__((address_space(3))) v4i_vec* lvec_t;
    {
        const char* gbase = reinterpret_cast<const char*>(X) +
                            (size_t)row0 * KDIM * sizeof(float);
        for (int i = tid; i < ROWS * (KDIM / 4); i += 256) {
            const int r  = i / (KDIM / 4);
            const int c4 = i % (KDIM / 4);
            if (row0 + r < Brows) {
                gvec_t g = (gvec_t)(uintptr_t)(gbase + (size_t)i * 16);
                lvec_t l = (lvec_t)(unsigned int)(uintptr_t)(&Xs[r * XS_STRIDE + c4 * 4]);
                __builtin_amdgcn_global_load_async_to_lds_b128(g, l, 0, 0);
            }
        }
    }
#else
    {
        const float4* X4 = reinterpret_cast<const float4*>(X + (size_t)row0 * KDIM);
        for (int i = tid; i < ROWS * (KDIM / 4); i += 256) {
            const int r  = i / (KDIM / 4);
            const int c4 = i % (KDIM / 4);
            float4 v = make_float4(0.f, 0.f, 0.f, 0.f);
            if (row0 + r < Brows) v = X4[i];
            *reinterpret_cast<float4*>(&Xs[r * XS_STRIDE + c4 * 4]) = v;
        }
    }
#endif
    __syncthreads();   // Ws zeroing complete (async Xs loads may still be in flight)

    // fill the two real rows of W_eff^T
    for (int i = tid; i < 2 * KDIM; i += 256) {
        const int n = i / KDIM;
        const int c = i % KDIM;
        Ws[n * WS_STRIDE + c] = Wc[i];
    }

#if USE_ASYNC_LDS
    // each wave drains its own async loads BEFORE the barrier, so after the
    // barrier every wave's LDS tile is guaranteed complete
    __builtin_amdgcn_s_wait_asynccnt(0);
#endif
    __syncthreads();

    // WMMA operand addressing (wave32):
    //  A 16x4 f32: lane l -> M = l&15, VGPR{0,1} = K = (l>>4)*2 + {0,1}
    //  B  4x16 f32: lane l -> N = l&15, VGPR{0,1} = K = (l>>4)*2 + {0,1}
    const int lane = tid & 31;
    const int wave = tid >> 5;                 // 0..7 -> 16-row tile
    const int m    = lane & 15;
    const int koff = (lane >> 4) * 2;
    const float* xrow = &Xs[(wave * 16 + m) * XS_STRIDE + koff];
    const float* wrow = &Ws[m * WS_STRIDE + koff];

    v8f acc = {0.f, 0.f, 0.f, 0.f, 0.f, 0.f, 0.f, 0.f};
#pragma unroll
    for (int k = 0; k < KDIM; k += 4) {
        v2f a = *reinterpret_cast<const v2f*>(&xrow[k]);
        v2f b = *reinterpret_cast<const v2f*>(&wrow[k]);
        acc = __builtin_amdgcn_wmma_f32_16x16x4_f32(
            /*neg_a=*/false, a, /*neg_b=*/false, b,
            /*c_mod=*/(short)0, acc, /*reuse_a=*/false, /*reuse_b=*/false);
    }

    // D 16x16 f32: VGPR v -> lanes 0-15: (M=v, N=lane); lanes 16-31: (M=v+8, N=lane-16)
    const int n     = (lane < 16) ? lane : (lane - 16);
    const int mbase = (lane < 16) ? 0 : 8;
    if (n < 2) {
        const int tile_row0 = row0 + wave * 16;
#pragma unroll
        for (int v = 0; v < 8; ++v) {
            const int r = tile_row0 + mbase + v;
            if (r < Brows) out[(size_t)r * 2 + n] = acc[v];
        }
    }
}

// ---------------------------------------------------------------------------
extern "C" void kernel_launch(void* const* d_in, const int* in_sizes, int n_in,
                              void* d_out, int out_size, void* d_ws, size_t ws_size,
                              hipStream_t stream) {
    const float* x     = (const float*)d_in[0];
    const float* coord = (const float*)d_in[1];
    const float* w1    = (const float*)d_in[2];
    const float* b1    = (const float*)d_in[3];
    const float* w2    = (const float*)d_in[4];
    const float* b2    = (const float*)d_in[5];
    const float* fc_w  = (const float*)d_in[6];
    float* out = (float*)d_out;
    float* Wc  = (float*)d_ws;                 // 400 floats of scratch

    const int Brows = in_sizes[0] / KDIM;      // 262144

    posegraph_prep_kernel<<<1, 256, 0, stream>>>(coord, w1, b1, w2, b2, fc_w, Wc);

    const int grid = (Brows + ROWS - 1) / ROWS;
    const size_t shmem = (size_t)(ROWS * XS_STRIDE + 16 * WS_STRIDE) * sizeof(float);
    posegraph_gemm_kernel<<<grid, 256, shmem, stream>>>(x, Wc, out, Brows);
}